// GraphSAGEBlock_83064667504981
// MI455X (gfx1250) — compile-verified
//
#include <hip/hip_runtime.h>
#include <hip/hip_bf16.h>

#define N_NODES  100000
#define N_EDGES  1600000
#define N_GRAPHS 128
#define DDIM     64
#define LN_EPS   1e-5f

typedef float v2f __attribute__((ext_vector_type(2)));
typedef float v8f __attribute__((ext_vector_type(8)));

__device__ __forceinline__ int imin(int a, int b) { return a < b ? a : b; }

// ---------------------------------------------------------------------------
// K1: zero workspace regions (msg_sum, deg, graph_accum)
// ---------------------------------------------------------------------------
__global__ void k_zero(float* p, int n) {
    int i = blockIdx.x * blockDim.x + threadIdx.x;
    if (i < n) p[i] = 0.0f;
}

// ---------------------------------------------------------------------------
// K2: edge scatter. One wave (32 lanes) per edge: lane reads float2 of
// x[src] row (fully coalesced 256B per wave) and atomically adds into
// msg_sum[dst]. Lane 0 increments deg[dst].
// ---------------------------------------------------------------------------
__global__ void k_scatter(const float* __restrict__ x,
                          const int*   __restrict__ eidx,
                          float* __restrict__ msg,
                          float* __restrict__ deg) {
    int gtid = blockIdx.x * blockDim.x + threadIdx.x;
    int edge = gtid >> 5;
    int lane = gtid & 31;
    if (edge >= N_EDGES) return;
    int src = eidx[edge];
    int dst = eidx[N_EDGES + edge];
    v2f v = *(const v2f*)(x + (long)src * DDIM + lane * 2);
    float* m = msg + (long)dst * DDIM + lane * 2;
    atomicAdd(m + 0, v.x);
    atomicAdd(m + 1, v.y);
    if (lane == 0) atomicAdd(deg + dst, 1.0f);
}

// ---------------------------------------------------------------------------
// K3: WMMA GEMM  node_pre = x @ W_self + (msg/deg) @ W_neigh + b
// plus fused graph pooling (segment-sum over sorted batch) via LDS.
// Block: 256 threads = 8 waves; each wave computes a 16x64 row tile.
// ---------------------------------------------------------------------------
__global__ void k_gemm_wmma(const float* __restrict__ x,
                            const float* __restrict__ msg,
                            const float* __restrict__ deg,
                            const int*   __restrict__ batch,
                            const float* __restrict__ Wself,
                            const float* __restrict__ Wneigh,
                            const float* __restrict__ bias,
                            float* __restrict__ node_pre,
                            float* __restrict__ gacc) {
    __shared__ float4 ldsW4[2048];              // 32 KB: W_self | W_neigh
    __shared__ float  ldsG[N_GRAPHS * DDIM];    // 32 KB graph accumulator
    float* lw = (float*)ldsW4;

    // stage weights into LDS (1024 float4 each)
    for (int i = threadIdx.x; i < 1024; i += 256) {
        ldsW4[i]        = ((const float4*)Wself)[i];
        ldsW4[i + 1024] = ((const float4*)Wneigh)[i];
    }
    for (int i = threadIdx.x; i < N_GRAPHS * DDIM; i += 256) ldsG[i] = 0.0f;
    __syncthreads();

    const int wid   = threadIdx.x >> 5;
    const int lane  = threadIdx.x & 31;
    const int mlane = lane & 15;         // M within tile for A, N-col for B
    const int khalf = lane >> 4;         // 0: K{0,1}  1: K{2,3}
    const int rbase = blockIdx.x * 128 + wid * 16;

    // A-row for this lane (clamped; out-of-range rows computed but not stored)
    const int arow = imin(rbase + mlane, N_NODES - 1);
    const float dg  = deg[arow];
    const float inv = 1.0f / fmaxf(dg, 1.0f);

    v8f acc[4];
    acc[0] = (v8f){}; acc[1] = (v8f){}; acc[2] = (v8f){}; acc[3] = (v8f){};

    const float* xrow = x   + (long)arow * DDIM;
    const float* mrow = msg + (long)arow * DDIM;

    for (int kk = 0; kk < 16; ++kk) {
        const int kb = kk * 4 + khalf * 2;
        v2f a_s = *(const v2f*)(xrow + kb);
        v2f a_n = *(const v2f*)(mrow + kb);
        a_n.x *= inv; a_n.y *= inv;
        #pragma unroll
        for (int nt = 0; nt < 4; ++nt) {
            const int ncol = nt * 16 + mlane;
            v2f b_s, b_n;
            b_s.x = lw[(kb + 0) * DDIM + ncol];
            b_s.y = lw[(kb + 1) * DDIM + ncol];
            b_n.x = lw[4096 + (kb + 0) * DDIM + ncol];
            b_n.y = lw[4096 + (kb + 1) * DDIM + ncol];
            acc[nt] = __builtin_amdgcn_wmma_f32_16x16x4_f32(
                false, a_s, false, b_s, (short)0, acc[nt], false, false);
            acc[nt] = __builtin_amdgcn_wmma_f32_16x16x4_f32(
                false, a_n, false, b_n, (short)0, acc[nt], false, false);
        }
    }

    // per-lane row ids for the C layout: element (v) -> M = v + 8*khalf
    int rowv[8], bidx[8];
    #pragma unroll
    for (int v = 0; v < 8; ++v) {
        rowv[v] = rbase + v + 8 * khalf;
        bidx[v] = batch[imin(rowv[v], N_NODES - 1)];
    }

    #pragma unroll
    for (int nt = 0; nt < 4; ++nt) {
        const int n  = nt * 16 + mlane;
        const float bn = bias[n];
        #pragma unroll
        for (int v = 0; v < 8; ++v) {
            float val = acc[nt][v] + bn;
            if (rowv[v] < N_NODES) {
                node_pre[(long)rowv[v] * DDIM + n] = val;
                atomicAdd(&ldsG[bidx[v] * DDIM + n], val);  // ds_add_f32
            }
        }
    }
    __syncthreads();

    // flush pooled block contribution; batch is sorted so the range is tiny
    const int bmn = batch[imin(blockIdx.x * 128, N_NODES - 1)];
    const int bmx = batch[imin(blockIdx.x * 128 + 127, N_NODES - 1)];
    const int span = (bmx - bmn + 1) * DDIM;
    for (int i = threadIdx.x; i < span; i += 256) {
        const int g = bmn + i / DDIM;
        const int d = i % DDIM;
        const float v = ldsG[g * DDIM + d];
        if (v != 0.0f) atomicAdd(&gacc[g * DDIM + d], v);
    }
}

// ---------------------------------------------------------------------------
// K4: per-node residual + LayerNorm + ReLU. One wave32 per row, 2 dims/lane.
// ---------------------------------------------------------------------------
__global__ void k_node_final(const float* __restrict__ node_pre,
                             const float* __restrict__ x,
                             const float* __restrict__ gamma,
                             const float* __restrict__ beta,
                             float* __restrict__ out) {
    const int gw   = (blockIdx.x * blockDim.x + threadIdx.x) >> 5;
    const int lane = threadIdx.x & 31;
    if (gw >= N_NODES) return;
    const long base = (long)gw * DDIM;
    float v0 = node_pre[base + lane]      + x[base + lane];
    float v1 = node_pre[base + lane + 32] + x[base + lane + 32];
    float s = v0 + v1, sq = v0 * v0 + v1 * v1;
    #pragma unroll
    for (int off = 16; off > 0; off >>= 1) {
        s  += __shfl_xor(s,  off, 32);
        sq += __shfl_xor(sq, off, 32);
    }
    const float mu  = s * (1.0f / 64.0f);
    const float var = sq * (1.0f / 64.0f) - mu * mu;
    const float rs  = rsqrtf(var + LN_EPS);
    float o0 = (v0 - mu) * rs * gamma[lane]      + beta[lane];
    float o1 = (v1 - mu) * rs * gamma[lane + 32] + beta[lane + 32];
    out[base + lane]      = fmaxf(o0, 0.0f);
    out[base + lane + 32] = fmaxf(o1, 0.0f);
}

// ---------------------------------------------------------------------------
// K5: per-graph residual + double LayerNorm. One wave32 per graph row.
// ---------------------------------------------------------------------------
__global__ void k_graph_final(const float* __restrict__ gacc,
                              const float* __restrict__ gembed,
                              const float* __restrict__ gamma,
                              const float* __restrict__ beta,
                              float* __restrict__ out) {
    const int g    = (blockIdx.x * blockDim.x + threadIdx.x) >> 5;
    const int lane = threadIdx.x & 31;
    if (g >= N_GRAPHS) return;
    const long base = (long)g * DDIM;
    float v0 = gacc[base + lane]      + gembed[base + lane];
    float v1 = gacc[base + lane + 32] + gembed[base + lane + 32];
    const float g0 = gamma[lane], g1 = gamma[lane + 32];
    const float b0 = beta[lane],  b1 = beta[lane + 32];
    #pragma unroll
    for (int pass = 0; pass < 2; ++pass) {
        float s = v0 + v1, sq = v0 * v0 + v1 * v1;
        #pragma unroll
        for (int off = 16; off > 0; off >>= 1) {
            s  += __shfl_xor(s,  off, 32);
            sq += __shfl_xor(sq, off, 32);
        }
        const float mu  = s * (1.0f / 64.0f);
        const float var = sq * (1.0f / 64.0f) - mu * mu;
        const float rs  = rsqrtf(var + LN_EPS);
        v0 = (v0 - mu) * rs * g0 + b0;
        v1 = (v1 - mu) * rs * g1 + b1;
    }
    out[base + lane]      = v0;
    out[base + lane + 32] = v1;
}

// ---------------------------------------------------------------------------
extern "C" void kernel_launch(void* const* d_in, const int* in_sizes, int n_in,
                              void* d_out, int out_size, void* d_ws, size_t ws_size,
                              hipStream_t stream) {
    const float* x      = (const float*)d_in[0];
    const int*   eidx   = (const int*)  d_in[1];
    const int*   batch  = (const int*)  d_in[2];
    const float* gembed = (const float*)d_in[3];
    const float* Wself  = (const float*)d_in[4];
    const float* Wneigh = (const float*)d_in[5];
    const float* bias   = (const float*)d_in[6];
    const float* gamma  = (const float*)d_in[7];
    const float* beta   = (const float*)d_in[8];
    float* out = (float*)d_out;

    // workspace layout (floats)
    float* msg      = (float*)d_ws;                       // N_NODES*DDIM
    float* deg      = msg + (size_t)N_NODES * DDIM;       // N_NODES
    float* gacc     = deg + N_NODES;                      // N_GRAPHS*DDIM
    float* node_pre = gacc + N_GRAPHS * DDIM;             // N_NODES*DDIM

    // K1: zero msg + deg + gacc (contiguous region)
    const int nzero = N_NODES * DDIM + N_NODES + N_GRAPHS * DDIM;
    k_zero<<<(nzero + 255) / 256, 256, 0, stream>>>(msg, nzero);

    // K2: edge scatter (one wave per edge)
    {
        const long threads = (long)N_EDGES * 32;
        k_scatter<<<(int)((threads + 255) / 256), 256, 0, stream>>>(x, eidx, msg, deg);
    }

    // K3: WMMA GEMM + fused pooling (128 rows per block)
    k_gemm_wmma<<<(N_NODES + 127) / 128, 256, 0, stream>>>(
        x, msg, deg, batch, Wself, Wneigh, bias, node_pre, gacc);

    // K4: node residual + LN + ReLU (one wave per node)
    {
        const long threads = (long)N_NODES * 32;
        k_node_final<<<(int)((threads + 255) / 256), 256, 0, stream>>>(
            node_pre, x, gamma, beta, out);
    }

    // K5: graph residual + LN(LN()) (one wave per graph)
    {
        const int threads = N_GRAPHS * 32;
        k_graph_final<<<(threads + 255) / 256, 256, 0, stream>>>(
            gacc, gembed, gamma, beta, out + (size_t)N_NODES * DDIM);
    }
}